// SymVectorBuild_78305843740823
// MI455X (gfx1250) — compile-verified
//
#include <hip/hip_runtime.h>
#include <hip/hip_bf16.h>

// ---------------- problem constants ----------------
#define NB   2      // batch
#define NC   256    // channels
#define NE   256    // embed
#define NS   3      // sample grid
#define NL   5      // levels
#define NN   128    // rois
#define HMX  128
#define KCK  (NC*9)     // 2304 im2col K
#define MCNV (NN*9)     // 1152 im2col M
#define EPSV 1e-5f

typedef __attribute__((ext_vector_type(2))) float v2f;
typedef __attribute__((ext_vector_type(8))) float v8f;

__device__ __forceinline__ v8f wmma4(v2f a, v2f b, v8f c) {
  // D = A(16x4,f32) * B(4x16,f32) + C(16x16,f32), full fp32 precision
  return __builtin_amdgcn_wmma_f32_16x16x4_f32(false, a, false, b, (short)0, c,
                                               false, false);
}

// ---------------- K1: ROI-align + im2col ----------------
// one block per roi (n), thread = channel c. Produces X[M=1152][K=2304].
__global__ void roi_im2col_kernel(const float* __restrict__ fm,
                                  const float* __restrict__ boxes,
                                  const int* __restrict__ batch_idx,
                                  const int* __restrict__ level_idx,
                                  float* __restrict__ X) {
  const int n = blockIdx.x;
  const int c = threadIdx.x;
  const int lvl = level_idx[n];
  const int b   = batch_idx[n];
  const float hv    = (float)(128 >> lvl);      // LEVEL_SIZES
  const float scale = 1.0f / (float)(8 << lvl); // 1/STRIDES

  const float x1 = boxes[n*4+0]*scale, y1 = boxes[n*4+1]*scale;
  const float x2 = boxes[n*4+2]*scale, y2 = boxes[n*4+3]*scale;
  const float rw = fmaxf(x2-x1, 1.0f), rh = fmaxf(y2-y1, 1.0f);

  const float* __restrict__ fmb =
      fm + ((((size_t)lvl*NB + b)*NC + c)*HMX)*HMX;

  float s[9];
  const int hi = (int)(hv - 1.0f);
  #pragma unroll
  for (int i = 0; i < 3; ++i) {
    const float y = y1 + ((float)i + 0.5f) * (1.0f/3.0f) * rh;
    #pragma unroll
    for (int j = 0; j < 3; ++j) {
      const float x = x1 + ((float)j + 0.5f) * (1.0f/3.0f) * rw;
      const bool valid = (y > -1.0f) & (y < hv) & (x > -1.0f) & (x < hv);
      const float yc = fminf(fmaxf(y, 0.0f), hv - 1.0f);
      const float xc = fminf(fmaxf(x, 0.0f), hv - 1.0f);
      const int y0 = (int)floorf(yc), x0 = (int)floorf(xc);
      const int y1i = min(y0 + 1, hi), x1i = min(x0 + 1, hi);
      const float ly = yc - (float)y0, lx = xc - (float)x0;
      const float v00 = fmb[y0 *HMX + x0 ];
      const float v01 = fmb[y0 *HMX + x1i];
      const float v10 = fmb[y1i*HMX + x0 ];
      const float v11 = fmb[y1i*HMX + x1i];
      float v = (1.0f-ly)*(1.0f-lx)*v00 + (1.0f-ly)*lx*v01
              + ly*(1.0f-lx)*v10       + ly*lx*v11;
      s[i*3+j] = valid ? v : 0.0f;
    }
  }

  // im2col for 3x3 conv, pad 1, 3x3 in -> 3x3 out
  #pragma unroll
  for (int oy = 0; oy < 3; ++oy)
    #pragma unroll
    for (int ox = 0; ox < 3; ++ox) {
      float* __restrict__ row = X + (size_t)(n*9 + oy*3 + ox)*KCK + c*9;
      #pragma unroll
      for (int ky = 0; ky < 3; ++ky)
        #pragma unroll
        for (int kx = 0; kx < 3; ++kx) {
          const int iy = oy + ky - 1, ix = ox + kx - 1;
          const bool inb = (iy >= 0) & (iy < 3) & (ix >= 0) & (ix < 3);
          row[ky*3 + kx] = inb ? s[iy*3 + ix] : 0.0f;
        }
    }
}

// ---------------- WMMA wave-tile GEMM core ----------------
// Out[m0:+16][e0:+64] += A[M][K] * Bt[N][K]^T, fp32 WMMA 16x16x4.
template <int K>
__device__ __forceinline__ void wave_gemm_16x64(const float* __restrict__ A,
                                                const float* __restrict__ Bt,
                                                int m0, int e0, int lane,
                                                v8f acc[4]) {
  const int half = lane >> 4;
  const int r    = lane & 15;
  const int koff = 2 * half;
  const float* __restrict__ arow = A  + (size_t)(m0 + r)      * K;
  const float* __restrict__ br0  = Bt + (size_t)(e0 +  0 + r) * K;
  const float* __restrict__ br1  = Bt + (size_t)(e0 + 16 + r) * K;
  const float* __restrict__ br2  = Bt + (size_t)(e0 + 32 + r) * K;
  const float* __restrict__ br3  = Bt + (size_t)(e0 + 48 + r) * K;
  #pragma unroll 4
  for (int k0 = 0; k0 < K; k0 += 4) {
    const v2f a  = *(const v2f*)(arow + k0 + koff);
    const v2f b0 = *(const v2f*)(br0  + k0 + koff);
    const v2f b1 = *(const v2f*)(br1  + k0 + koff);
    const v2f b2 = *(const v2f*)(br2  + k0 + koff);
    const v2f b3 = *(const v2f*)(br3  + k0 + koff);
    acc[0] = wmma4(a, b0, acc[0]);
    acc[1] = wmma4(a, b1, acc[1]);
    acc[2] = wmma4(a, b2, acc[2]);
    acc[3] = wmma4(a, b3, acc[3]);
  }
}

// ---------------- K2: conv-as-GEMM (M=1152,K=2304,N=256) ----------------
// Bt = conv_w, already [E][C*9] row-major. 288 waves exactly (no partial waves).
__global__ void conv_gemm_kernel(const float* __restrict__ X,
                                 const float* __restrict__ conv_w,
                                 float* __restrict__ Out1) {
  const int wave = blockIdx.x * (blockDim.x >> 5) + (threadIdx.x >> 5);
  const int lane = threadIdx.x & 31;
  const int NT = NE / 64;          // 4
  const int tm = wave / NT;        // 0..71
  const int tn = wave % NT;
  const int m0 = tm * 16, e0 = tn * 64;
  v8f acc[4] = {v8f{}, v8f{}, v8f{}, v8f{}};
  wave_gemm_16x64<KCK>(X, conv_w, m0, e0, lane, acc);
  const int half = lane >> 4, r = lane & 15;
  #pragma unroll
  for (int t = 0; t < 4; ++t)
    #pragma unroll
    for (int i = 0; i < 8; ++i)
      Out1[(size_t)(m0 + i + 8*half)*NE + e0 + t*16 + r] = acc[t][i];
}

// ---------------- K3: GroupNorm(8) + ReLU + mean-pool ----------------
// one block per roi; wave32 == one group of 32 channels -> pure shuffle reduce.
__global__ void gn_pool_kernel(const float* __restrict__ Out1,
                               const float* __restrict__ gn_scale,
                               const float* __restrict__ gn_bias,
                               float* __restrict__ pooled) {
  const int n = blockIdx.x;
  const int e = threadIdx.x;   // 0..255; lanes [32g,32g+32) form group g
  float v[9];
  float s1 = 0.0f, s2 = 0.0f;
  #pragma unroll
  for (int p = 0; p < 9; ++p) {
    v[p] = Out1[(size_t)(n*9 + p)*NE + e];
    s1 += v[p];
    s2 += v[p]*v[p];
  }
  #pragma unroll
  for (int off = 16; off > 0; off >>= 1) {
    s1 += __shfl_xor(s1, off, 32);
    s2 += __shfl_xor(s2, off, 32);
  }
  const float mean = s1 * (1.0f/288.0f);
  const float var  = s2 * (1.0f/288.0f) - mean*mean;
  const float inv  = rsqrtf(var + EPSV);
  const float g = gn_scale[e], bb = gn_bias[e];
  float acc = 0.0f;
  #pragma unroll
  for (int p = 0; p < 9; ++p)
    acc += fmaxf((v[p] - mean)*inv*g + bb, 0.0f);
  pooled[(size_t)n*NE + e] = acc * (1.0f/9.0f);
}

// ---------------- K4: head GEMM + spat + sym_emb fuse-in ----------------
// M=128,K=256,N=256; Bt = head_w [f][e] row-major. 32 waves exactly.
__global__ void head_gemm_kernel(const float* __restrict__ pooled,
                                 const float* __restrict__ head_w,
                                 const float* __restrict__ head_b,
                                 const float* __restrict__ boxes,
                                 const float* __restrict__ spat_w,
                                 const float* __restrict__ spat_b,
                                 const float* __restrict__ sym_emb,
                                 const int* __restrict__ sym_ids,
                                 float* __restrict__ fused) {
  const int wave = blockIdx.x * (blockDim.x >> 5) + (threadIdx.x >> 5);
  const int lane = threadIdx.x & 31;
  const int NT = NE / 64;          // 4
  const int tm = wave / NT;        // 0..7
  const int tn = wave % NT;
  const int m0 = tm * 16, e0 = tn * 64;
  v8f acc[4] = {v8f{}, v8f{}, v8f{}, v8f{}};
  wave_gemm_16x64<NE>(pooled, head_w, m0, e0, lane, acc);
  const int half = lane >> 4, r = lane & 15;
  #pragma unroll
  for (int t = 0; t < 4; ++t) {
    const int f = e0 + t*16 + r;
    const float hb = head_b[f];
    const float sw0 = spat_w[f*4+0], sw1 = spat_w[f*4+1];
    const float sw2 = spat_w[f*4+2], sw3 = spat_w[f*4+3];
    const float sb = spat_b[f];
    #pragma unroll
    for (int i = 0; i < 8; ++i) {
      const int nn = m0 + i + 8*half;
      float val = fmaxf(acc[t][i] + hb, 0.0f);                 // head relu
      float sp = sb + boxes[nn*4+0]*sw0 + boxes[nn*4+1]*sw1
                    + boxes[nn*4+2]*sw2 + boxes[nn*4+3]*sw3;
      sp = fmaxf(sp, 0.0f);                                    // spat relu
      val += sp + sym_emb[(size_t)sym_ids[nn]*NE + f];
      fused[(size_t)nn*NE + f] = val;
    }
  }
}

// ---------------- K5: per-row fuse_w[sym] matvec + bias + ReLU ----------------
__global__ void fuse_mv_kernel(const float* __restrict__ fused,
                               const float* __restrict__ fuse_w,
                               const float* __restrict__ fuse_b,
                               const int* __restrict__ sym_ids,
                               float* __restrict__ out) {
  __shared__ float fin[NE];
  const int n = blockIdx.x, f = threadIdx.x;
  fin[f] = fused[(size_t)n*NE + f];
  __syncthreads();
  const int s = sym_ids[n];
  const float* __restrict__ w = fuse_w + ((size_t)s*NE + f)*NE;
  float acc = fuse_b[(size_t)s*NE + f];
  #pragma unroll 8
  for (int e = 0; e < NE; ++e)
    acc = fmaf(fin[e], w[e], acc);
  out[(size_t)n*NE + f] = fmaxf(acc, 0.0f);
}

// ---------------- launcher ----------------
extern "C" void kernel_launch(void* const* d_in, const int* in_sizes, int n_in,
                              void* d_out, int out_size, void* d_ws, size_t ws_size,
                              hipStream_t stream) {
  const float* fm        = (const float*)d_in[0];
  const float* boxes     = (const float*)d_in[1];
  const float* conv_w    = (const float*)d_in[2];
  const float* gn_scale  = (const float*)d_in[3];
  const float* gn_bias   = (const float*)d_in[4];
  const float* head_w    = (const float*)d_in[5];
  const float* head_b    = (const float*)d_in[6];
  const float* spat_w    = (const float*)d_in[7];
  const float* spat_b    = (const float*)d_in[8];
  const float* sym_emb   = (const float*)d_in[9];
  const float* fuse_w    = (const float*)d_in[10];
  const float* fuse_b    = (const float*)d_in[11];
  const int*   batch_idx = (const int*)d_in[12];
  const int*   level_idx = (const int*)d_in[13];
  const int*   sym_ids   = (const int*)d_in[14];
  float* out = (float*)d_out;

  float* ws     = (float*)d_ws;
  float* X      = ws;                       // 1152*2304
  float* Out1   = X    + (size_t)MCNV*KCK;  // 1152*256
  float* pooled = Out1 + (size_t)MCNV*NE;   // 128*256
  float* fused  = pooled + (size_t)NN*NE;   // 128*256

  roi_im2col_kernel<<<NN, NC, 0, stream>>>(fm, boxes, batch_idx, level_idx, X);
  // conv GEMM: 72 M-tiles * 4 N-tiles = 288 waves = 36 blocks of 8 waves
  conv_gemm_kernel<<<36, 256, 0, stream>>>(X, conv_w, Out1);
  gn_pool_kernel<<<NN, NE, 0, stream>>>(Out1, gn_scale, gn_bias, pooled);
  // head GEMM: 8 * 4 = 32 waves = 4 blocks of 8 waves
  head_gemm_kernel<<<4, 256, 0, stream>>>(pooled, head_w, head_b, boxes,
                                          spat_w, spat_b, sym_emb, sym_ids, fused);
  fuse_mv_kernel<<<NN, NE, 0, stream>>>(fused, fuse_w, fuse_b, sym_ids, out);
}